// InvariantMapping_8899172238145
// MI455X (gfx1250) — compile-verified
//
#include <hip/hip_runtime.h>
#include <stdint.h>

// Problem constants (from reference: b=16, c=256, d=3, n=16384, fp32).
namespace {
constexpr int kB = 16;
constexpr int kC = 256;
constexpr int kD = 3;
constexpr int kN = 16384;

constexpr int kW = 256;                    // n-columns handled per block (1 per thread)
constexpr int kCC = 4;                     // channels per LDS chunk
constexpr int kChunks = kC / kCC;          // 64 chunks
constexpr int kRows = kCC * 2 * kD;        // 24 rows per chunk (fx d0..2, fy d0..2) x kCC
constexpr int kSegs = kRows * (kW / 4);    // 1536 16-byte segments per chunk
constexpr int kIssues = kSegs / 256;       // 6 async b128 issues per thread per chunk
constexpr int kBufBytes = kRows * kW * 4;  // 24576 bytes per buffer (x2 = 48KB LDS)
constexpr float kEps = 1e-6f;
}  // namespace

// -------------------------------------------------------------------------
// Kernel 1: streaming reduction. Computes per-(b,n):
//   Sx[3], Sy[3] (channel sums) and M[3][3] = sum_c fx_d * fy_d'
// using async global->LDS B128 copies (ASYNCcnt double buffering), then the
// normalized score s[b,n] = px . M . py, matching the reference numerics.
// -------------------------------------------------------------------------
__global__ __launch_bounds__(256) void score_kernel(const float* __restrict__ fx,
                                                    const float* __restrict__ fy,
                                                    float* __restrict__ scores) {
  __shared__ __align__(16) float buf[2][kRows][kW];

  const int t = threadIdx.x;
  const int n0 = blockIdx.x * kW;
  const int b = blockIdx.y;

  // Low 32 bits of a generic pointer to LDS == wave-relative LDS byte offset.
  const unsigned lds0 = (unsigned)(uintptr_t)(&buf[0][0][0]);

  // Per-thread constant assignment of 6 x 16-byte segments per chunk.
  const float* ga[kIssues];  // global source address (chunk 0)
  unsigned la[kIssues];      // LDS dest byte address (buffer 0)
#pragma unroll
  for (int i = 0; i < kIssues; ++i) {
    const int seg = i * 256 + t;     // 0..1535
    const int row = seg >> 6;        // 0..23 : row = cc*6 + q
    const int col = seg & 63;        // 16-byte column within 1KB row
    const int cc = row / 6;
    const int q = row % 6;           // 0..2 -> fx d, 3..5 -> fy d
    const int d = (q < 3) ? q : q - 3;
    const float* base = (q < 3) ? fx : fy;
    const size_t eoff =
        ((((size_t)b * kC + (size_t)cc) * kD + (size_t)d) * kN) + (size_t)n0 + (size_t)col * 4;
    ga[i] = base + eoff;
    la[i] = lds0 + (unsigned)seg * 16u;
  }
  const size_t kChunkStride = (size_t)kCC * kD * kN;  // element advance per chunk (c += 4)

  // Prologue: issue chunk 0 into buffer 0.
#pragma unroll
  for (int i = 0; i < kIssues; ++i) {
    asm volatile("global_load_async_to_lds_b128 %0, %1, off" ::"v"(la[i]), "v"(ga[i])
                 : "memory");
  }

  float Sx0 = 0.f, Sx1 = 0.f, Sx2 = 0.f;
  float Sy0 = 0.f, Sy1 = 0.f, Sy2 = 0.f;
  float M00 = 0.f, M01 = 0.f, M02 = 0.f;
  float M10 = 0.f, M11 = 0.f, M12 = 0.f;
  float M20 = 0.f, M21 = 0.f, M22 = 0.f;

  unsigned nextoff = (unsigned)kBufBytes;  // LDS buffer offset for the *next* chunk

  for (int k = 0; k < kChunks; ++k) {
    const int h = k & 1;
    if (k + 1 < kChunks) {
      // Issue chunk k+1 into the other buffer, then retire chunk k (oldest 6).
#pragma unroll
      for (int i = 0; i < kIssues; ++i) {
        ga[i] += kChunkStride;
        const unsigned l = la[i] + nextoff;
        asm volatile("global_load_async_to_lds_b128 %0, %1, off" ::"v"(l), "v"(ga[i])
                     : "memory");
      }
      nextoff ^= (unsigned)kBufBytes;
      asm volatile("s_wait_asynccnt 6" ::: "memory");
    } else {
      asm volatile("s_wait_asynccnt 0" ::: "memory");
    }
    __syncthreads();  // all waves' chunk-k data visible in LDS

#pragma unroll
    for (int cc = 0; cc < kCC; ++cc) {
      const float x0 = buf[h][cc * 6 + 0][t];
      const float x1 = buf[h][cc * 6 + 1][t];
      const float x2 = buf[h][cc * 6 + 2][t];
      const float y0 = buf[h][cc * 6 + 3][t];
      const float y1 = buf[h][cc * 6 + 4][t];
      const float y2 = buf[h][cc * 6 + 5][t];
      Sx0 += x0; Sx1 += x1; Sx2 += x2;
      Sy0 += y0; Sy1 += y1; Sy2 += y2;
      M00 = fmaf(x0, y0, M00); M01 = fmaf(x0, y1, M01); M02 = fmaf(x0, y2, M02);
      M10 = fmaf(x1, y0, M10); M11 = fmaf(x1, y1, M11); M12 = fmaf(x1, y2, M12);
      M20 = fmaf(x2, y0, M20); M21 = fmaf(x2, y1, M21); M22 = fmaf(x2, y2, M22);
    }
    __syncthreads();  // buffer h free for reuse (WAR guard for chunk k+2)
  }

  // Epilogue: reproduce reference normalization numerics.
  const float inv = 1.0f / (float)kC;
  const float mx0 = Sx0 * inv, mx1 = Sx1 * inv, mx2 = Sx2 * inv;
  const float my0 = Sy0 * inv, my1 = Sy1 * inv, my2 = Sy2 * inv;
  const float nx = sqrtf(mx0 * mx0 + mx1 * mx1 + mx2 * mx2) + kEps;
  const float ny = sqrtf(my0 * my0 + my1 * my1 + my2 * my2) + kEps;
  const float px0 = mx0 / nx, px1 = mx1 / nx, px2 = mx2 / nx;
  const float py0 = my0 / ny, py1 = my1 / ny, py2 = my2 / ny;
  const float s = px0 * (M00 * py0 + M01 * py1 + M02 * py2) +
                  px1 * (M10 * py0 + M11 * py1 + M12 * py2) +
                  px2 * (M20 * py0 + M21 * py1 + M22 * py2);
  scores[(size_t)b * kN + (size_t)(n0 + t)] = s;
}

// -------------------------------------------------------------------------
// Kernel 2: top-K indices per batch, matching jax.lax.top_k order
// (descending value, ties -> smaller index first). One block per b.
// Key = (monotone float bits << 32) | (0xFFFFFFFF - index): larger = better.
// -------------------------------------------------------------------------
__global__ __launch_bounds__(256) void topk_kernel(const float* __restrict__ scores,
                                                   int* __restrict__ idx, int K) {
  __shared__ unsigned long long red[256];
  const int b = blockIdx.x;
  const int t = threadIdx.x;
  const float* s = scores + (size_t)b * kN;
  unsigned long long sel[16];

  for (int j = 0; j < K; ++j) {
    unsigned long long best = 0ull;
    for (int i = t; i < kN; i += 256) {
      unsigned u = __float_as_uint(s[i]);
      u = (u & 0x80000000u) ? ~u : (u | 0x80000000u);  // monotone order map
      const unsigned long long key =
          ((unsigned long long)u << 32) | (unsigned)(0xFFFFFFFFu - (unsigned)i);
      bool taken = false;
      for (int p = 0; p < j; ++p) taken = taken || (sel[p] == key);
      if (!taken && key > best) best = key;
    }
    red[t] = best;
    __syncthreads();
    for (int off = 128; off > 0; off >>= 1) {
      if (t < off) {
        const unsigned long long o = red[t + off];
        if (o > red[t]) red[t] = o;
      }
      __syncthreads();
    }
    const unsigned long long win = red[0];
    sel[j] = win;
    if (t == 0) idx[b * K + j] = (int)(0xFFFFFFFFu - (unsigned)(win & 0xFFFFFFFFu));
    __syncthreads();
  }
}

// -------------------------------------------------------------------------
// Kernel 3: gather selected columns. d_out = [fx_sel | fy_sel] flat.
// fx_sel[b,c,d,j] = fx[b,c,d, idx[b,j]]
// -------------------------------------------------------------------------
__global__ __launch_bounds__(256) void gather_kernel(const float* __restrict__ fx,
                                                     const float* __restrict__ fy,
                                                     const int* __restrict__ idx,
                                                     float* __restrict__ out, int K) {
  const int g = blockIdx.x * 256 + threadIdx.x;
  const int half_sz = kB * kC * kD * K;
  if (g >= 2 * half_sz) return;
  const float* src = (g < half_sz) ? fx : fy;
  const int r = (g < half_sz) ? g : g - half_sz;
  const int j = r % K;
  const int rem = r / K;  // = (b*kC + c)*kD + d
  const int b = rem / (kC * kD);
  const int n = idx[b * K + j];
  out[g] = src[(size_t)rem * kN + (size_t)n];
}

extern "C" void kernel_launch(void* const* d_in, const int* in_sizes, int n_in,
                              void* d_out, int out_size, void* d_ws, size_t ws_size,
                              hipStream_t stream) {
  const float* fx = (const float*)d_in[0];
  const float* fy = (const float*)d_in[1];
  float* out = (float*)d_out;

  // Workspace layout: [scores: kB*kN floats][idx: kB*K ints]
  float* scores = (float*)d_ws;
  int* idx = (int*)(scores + (size_t)kB * kN);

  // k = b // topk, derived deterministically from the output size:
  // out_size = 2 * kB * kC * kD * k
  int K = out_size / (2 * kB * kC * kD);
  if (K < 1) K = 1;
  if (K > 16) K = 16;

  dim3 grid1(kN / kW, kB);
  score_kernel<<<grid1, 256, 0, stream>>>(fx, fy, scores);

  topk_kernel<<<kB, 256, 0, stream>>>(scores, idx, K);

  const int total = 2 * kB * kC * kD * K;
  gather_kernel<<<(total + 255) / 256, 256, 0, stream>>>(fx, fy, idx, out, K);
}